// SparseLDS_56822417326795
// MI455X (gfx1250) — compile-verified
//
#include <hip/hip_runtime.h>
#include <math.h>

typedef float v2f __attribute__((ext_vector_type(2)));
typedef float v8f __attribute__((ext_vector_type(8)));

#define T_LEN 2048
#define B_DIM 16
#define K_DIM 64
#define N_DIM 32
#define M_DIM 32

// One workgroup per k. 4 waves (wave32) split the T dimension.
// G[b,n] = sum_u x[u,b,k] * lam[k,n]^(T-2-u)  via V_WMMA_F32_16X16X4_F32,
// weights advanced by a decaying complex recurrence W(u-4) = W(u)*lam^4.
// Then y_k[b,m] = sum_n G[b,n]*(Bp[n]*C[n,m]) + x[T-2,b,k]*D[k,m] + Do[k,m].
__global__ __launch_bounds__(128)
void SparseLDS_scan_wmma(const float* __restrict__ x,
                         const float* __restrict__ lnlam_re,
                         const float* __restrict__ lnlam_im,
                         const float* __restrict__ C_re,
                         const float* __restrict__ C_im,
                         const float* __restrict__ Dm,
                         const float* __restrict__ Dom,
                         float* __restrict__ ws_y)   // [K][B][M][2] re/im
{
    const int k    = blockIdx.x;
    const int tid  = threadIdx.x;
    const int wave = tid >> 5;
    const int lane = tid & 31;
    const int col  = lane & 15;
    const bool lo  = lane < 16;

    __shared__ float s_lnre[N_DIM], s_lnim[N_DIM];
    __shared__ float s_bpre[N_DIM], s_bpim[N_DIM];
    __shared__ float s_cqre[N_DIM][M_DIM], s_cqim[N_DIM][M_DIM];
    __shared__ float s_part[4][2][B_DIM][N_DIM];   // per-wave partials re/im
    __shared__ float s_gre[B_DIM][N_DIM], s_gim[B_DIM][N_DIM];

    // ---- Stage 1: load ln(lambda) for this k ----
    if (tid < N_DIM) {
        s_lnre[tid] = lnlam_re[k * N_DIM + tid];
        s_lnim[tid] = lnlam_im[k * N_DIM + tid];
    }
    __syncthreads();

    // ---- Stage 2: partial-fraction Bp[j] = exp(-sum_{i!=j} log(1 - lam_i/lam_j)) ----
    if (tid < N_DIM) {
        const int j = tid;
        const float ljre = s_lnre[j], ljim = s_lnim[j];
        float sre = 0.f, sim = 0.f;
        for (int i = 0; i < N_DIM; ++i) {
            if (i == j) continue;
            float dre = s_lnre[i] - ljre;
            float dim = s_lnim[i] - ljim;
            float mag = expf(dre);
            float rre = mag * cosf(dim);
            float rim = mag * sinf(dim);
            float wre = 1.f - rre;
            float wim = -rim;
            sre += 0.5f * logf(wre * wre + wim * wim);
            sim += atan2f(wim, wre);
        }
        float be = expf(-sre);
        s_bpre[j] = be * cosf(-sim);
        s_bpim[j] = be * sinf(-sim);
    }
    __syncthreads();

    // ---- Stage 3: fold Bp into C: Cq[n][m] = Bp[n] * (C_re + i C_im) ----
    for (int idx = tid; idx < N_DIM * M_DIM; idx += 128) {
        int n = idx >> 5, m = idx & 31;
        float cre = C_re[k * N_DIM * M_DIM + idx];
        float cim = C_im[k * N_DIM * M_DIM + idx];
        float bre = s_bpre[n], bim = s_bpim[n];
        s_cqre[n][m] = bre * cre - bim * cim;
        s_cqim[n][m] = bre * cim + bim * cre;
    }

    // ---- Stage 4: WMMA time-reduction with recurrent weights ----
    // A-frag (16x4 f32): lanes 0-15 hold (t+0,t+1) for b=col; lanes 16-31 hold (t+2,t+3).
    // B-frag (4x16 f32): same per-lane t-offsets, n = col (first half) / col+16 (second).
    const int n0 = col, n1 = col + 16;
    const float lr0 = s_lnre[n0], li0 = s_lnim[n0];
    const float lr1 = s_lnre[n1], li1 = s_lnim[n1];

    // lam^4 (decaying, |.| < 1) for the downward recurrence
    float l4r0, l4i0, l4r1, l4i1;
    {
        float e = expf(4.f * lr0); l4r0 = e * cosf(4.f * li0); l4i0 = e * sinf(4.f * li0);
        e       = expf(4.f * lr1); l4r1 = e * cosf(4.f * li1); l4i1 = e * sinf(4.f * li1);
    }

    const int toff0 = lo ? 0 : 2;
    const int toff1 = lo ? 1 : 3;
    const int ubeg  = wave * (T_LEN / 4);
    const int utop  = ubeg + (T_LEN / 4) - 4;

    // initial weights at the top chunk: W = lam^(T-2-u) (true powers; the u=T-1
    // shift-mask is applied to the A-fragment in the peeled first iteration)
    const float p0 = (float)(T_LEN - 2 - (utop + toff0));
    const float p1 = (float)(T_LEN - 2 - (utop + toff1));
    float w00r, w00i, w10r, w10i, w01r, w01i, w11r, w11i;
    { float e = expf(p0 * lr0); w00r = e * cosf(p0 * li0); w00i = e * sinf(p0 * li0); }
    { float e = expf(p1 * lr0); w10r = e * cosf(p1 * li0); w10i = e * sinf(p1 * li0); }
    { float e = expf(p0 * lr1); w01r = e * cosf(p0 * li1); w01i = e * sinf(p0 * li1); }
    { float e = expf(p1 * lr1); w11r = e * cosf(p1 * li1); w11i = e * sinf(p1 * li1); }

    v8f acc_re0 = {}; v8f acc_im0 = {};
    v8f acc_re1 = {}; v8f acc_im1 = {};

    const float* xk = x + col * K_DIM + k;
    const float* px0 = xk + (size_t)(utop + toff0) * (B_DIM * K_DIM);
    const float* px1 = xk + (size_t)(utop + toff1) * (B_DIM * K_DIM);

    // ---- peeled first iteration (only place the u = T-1 mask can fire) ----
    {
        v2f a;
        a.x = px0[0];
        float ay = px1[0];                          // in-range load, then select
        if ((utop + toff1) == T_LEN - 1) ay = 0.f;  // -> v_cndmask, no branch
        a.y = ay;

        v2f bre0; bre0.x = w00r; bre0.y = w10r;
        v2f bim0; bim0.x = w00i; bim0.y = w10i;
        v2f bre1; bre1.x = w01r; bre1.y = w11r;
        v2f bim1; bim1.x = w01i; bim1.y = w11i;

        acc_re0 = __builtin_amdgcn_wmma_f32_16x16x4_f32(false, a, false, bre0,
                                                        (short)0, acc_re0, false, false);
        acc_im0 = __builtin_amdgcn_wmma_f32_16x16x4_f32(false, a, false, bim0,
                                                        (short)0, acc_im0, false, false);
        acc_re1 = __builtin_amdgcn_wmma_f32_16x16x4_f32(false, a, false, bre1,
                                                        (short)0, acc_re1, false, false);
        acc_im1 = __builtin_amdgcn_wmma_f32_16x16x4_f32(false, a, false, bim1,
                                                        (short)0, acc_im1, false, false);

        float t;
        t = w00r * l4r0 - w00i * l4i0; w00i = w00r * l4i0 + w00i * l4r0; w00r = t;
        t = w10r * l4r0 - w10i * l4i0; w10i = w10r * l4i0 + w10i * l4r0; w10r = t;
        t = w01r * l4r1 - w01i * l4i1; w01i = w01r * l4i1 + w01i * l4r1; w01r = t;
        t = w11r * l4r1 - w11i * l4i1; w11i = w11r * l4i1 + w11i * l4r1; w11r = t;

        px0 -= 4 * (B_DIM * K_DIM);
        px1 -= 4 * (B_DIM * K_DIM);
    }

    // ---- mask-free steady-state loop: 127 iterations ----
    for (int it = 0; it < (T_LEN / 16) - 1; ++it) {
        v2f a;
        a.x = px0[0];
        a.y = px1[0];

        v2f bre0; bre0.x = w00r; bre0.y = w10r;
        v2f bim0; bim0.x = w00i; bim0.y = w10i;
        v2f bre1; bre1.x = w01r; bre1.y = w11r;
        v2f bim1; bim1.x = w01i; bim1.y = w11i;

        acc_re0 = __builtin_amdgcn_wmma_f32_16x16x4_f32(false, a, false, bre0,
                                                        (short)0, acc_re0, false, false);
        acc_im0 = __builtin_amdgcn_wmma_f32_16x16x4_f32(false, a, false, bim0,
                                                        (short)0, acc_im0, false, false);
        acc_re1 = __builtin_amdgcn_wmma_f32_16x16x4_f32(false, a, false, bre1,
                                                        (short)0, acc_re1, false, false);
        acc_im1 = __builtin_amdgcn_wmma_f32_16x16x4_f32(false, a, false, bim1,
                                                        (short)0, acc_im1, false, false);

        // advance weights: W(u-4) = W(u) * lam^4   (4 complex multiplies)
        float t;
        t = w00r * l4r0 - w00i * l4i0; w00i = w00r * l4i0 + w00i * l4r0; w00r = t;
        t = w10r * l4r0 - w10i * l4i0; w10i = w10r * l4i0 + w10i * l4r0; w10r = t;
        t = w01r * l4r1 - w01i * l4i1; w01i = w01r * l4i1 + w01i * l4r1; w01r = t;
        t = w11r * l4r1 - w11i * l4i1; w11i = w11r * l4i1 + w11i * l4r1; w11r = t;

        px0 -= 4 * (B_DIM * K_DIM);
        px1 -= 4 * (B_DIM * K_DIM);
    }

    // ---- Stage 5: spill per-wave accumulators (C/D layout: b = i + (lo?0:8)) ----
    {
        const int brow = lo ? 0 : 8;
        for (int i = 0; i < 8; ++i) {
            s_part[wave][0][brow + i][n0] = acc_re0[i];
            s_part[wave][1][brow + i][n0] = acc_im0[i];
            s_part[wave][0][brow + i][n1] = acc_re1[i];
            s_part[wave][1][brow + i][n1] = acc_im1[i];
        }
    }
    __syncthreads();

    // ---- Stage 6: deterministic cross-wave reduction -> G ----
    for (int idx = tid; idx < 2 * B_DIM * N_DIM; idx += 128) {
        int comp = idx >> 9;            // 512 = B_DIM*N_DIM
        int rem  = idx & 511;
        int b = rem >> 5, n = rem & 31;
        float s = s_part[0][comp][b][n] + s_part[1][comp][b][n]
                + s_part[2][comp][b][n] + s_part[3][comp][b][n];
        if (comp == 0) s_gre[b][n] = s; else s_gim[b][n] = s;
    }
    __syncthreads();

    // ---- Stage 7: y_k[b,m] = sum_n G[b,n]*Cq[n,m] + x[T-2,b,k]*D[k,m] + Do[k,m] ----
    for (int idx = tid; idx < B_DIM * M_DIM; idx += 128) {
        int b = idx >> 5, m = idx & 31;
        float yr = 0.f, yi = 0.f;
        for (int n = 0; n < N_DIM; ++n) {
            float gr = s_gre[b][n], gi = s_gim[b][n];
            float cr = s_cqre[n][m], ci = s_cqim[n][m];
            yr += gr * cr - gi * ci;
            yi += gr * ci + gi * cr;
        }
        float xlast = x[(T_LEN - 2) * (B_DIM * K_DIM) + b * K_DIM + k];
        yr += xlast * Dm[k * M_DIM + m] + Dom[k * M_DIM + m];
        float* o = ws_y + (((size_t)k * B_DIM + b) * M_DIM + m) * 2;
        o[0] = yr;
        o[1] = yi;
    }
}

// Mean over k; writes interleaved complex64 (B, M).
__global__ __launch_bounds__(256)
void SparseLDS_reduce_k(const float* __restrict__ ws_y, float* __restrict__ out)
{
    int bm = blockIdx.x * blockDim.x + threadIdx.x;
    if (bm >= B_DIM * M_DIM) return;
    float sr = 0.f, si = 0.f;
    for (int k = 0; k < K_DIM; ++k) {
        const float* p = ws_y + ((size_t)k * (B_DIM * M_DIM) + bm) * 2;
        sr += p[0];
        si += p[1];
    }
    out[2 * bm + 0] = sr * (1.0f / K_DIM);
    out[2 * bm + 1] = si * (1.0f / K_DIM);
}

extern "C" void kernel_launch(void* const* d_in, const int* in_sizes, int n_in,
                              void* d_out, int out_size, void* d_ws, size_t ws_size,
                              hipStream_t stream) {
    const float* x    = (const float*)d_in[0];
    const float* lnre = (const float*)d_in[1];
    const float* lnim = (const float*)d_in[2];
    const float* Cre  = (const float*)d_in[3];
    const float* Cim  = (const float*)d_in[4];
    const float* Dm   = (const float*)d_in[5];
    const float* Dom  = (const float*)d_in[6];
    float* wsy = (float*)d_ws;   // needs K*B*M*2*4 = 256 KB

    SparseLDS_scan_wmma<<<K_DIM, 128, 0, stream>>>(x, lnre, lnim, Cre, Cim, Dm, Dom, wsy);
    SparseLDS_reduce_k<<<(B_DIM * M_DIM + 255) / 256, 256, 0, stream>>>(wsy, (float*)d_out);
}